// NR_GraphAttention_51831665328618
// MI455X (gfx1250) — compile-verified
//
#include <hip/hip_runtime.h>

typedef __attribute__((ext_vector_type(2))) float v2f;
typedef __attribute__((ext_vector_type(4))) float v4f;
typedef __attribute__((ext_vector_type(8))) float v8f;

#define DDIM   128   // feature dim
#define DEG    16    // edges per node (contiguous)
#define ROWP   132   // padded LDS row stride in dwords (bank-conflict-free fragments)
#define WPB    4     // waves (= nodes) per block

// Wave-uniform broadcasts via v_readlane (SGPR result; lane index is constant).
static __device__ __forceinline__ int bcast_i(int v, int lane) {
    return __builtin_amdgcn_readlane(v, lane);
}
static __device__ __forceinline__ float bcast_f(float v, int lane) {
    return __uint_as_float((unsigned)__builtin_amdgcn_readlane((int)__float_as_uint(v), lane));
}

// ---------------------------------------------------------------------------
// Pre-kernel: per-relation attention scalar and inverse norm (2000 rows only)
//   attv[r] = exp(<rel_emb[r], attn_kernel>)
//   innv[r] = rsqrt(max(||rel_emb[r]||^2, 1e-12))
// ---------------------------------------------------------------------------
__global__ __launch_bounds__(256) void relpre_kernel(
    const float* __restrict__ rel_emb,
    const float* __restrict__ attn_kernel,
    float* __restrict__ attv,
    float* __restrict__ innv,
    int n_rels)
{
    const int lane = threadIdx.x & 31;
    const int r = (int)((blockIdx.x * blockDim.x + threadIdx.x) >> 5);
    if (r >= n_rels) return;
    const float4 rv = *(const float4*)(rel_emb + (size_t)r * DDIM + lane * 4);
    const float4 kv = *(const float4*)(attn_kernel + lane * 4);
    float d = rv.x * kv.x + rv.y * kv.y + rv.z * kv.z + rv.w * kv.w;
    float n = rv.x * rv.x + rv.y * rv.y + rv.z * rv.z + rv.w * rv.w;
    #pragma unroll
    for (int off = 16; off >= 1; off >>= 1) {
        d += __shfl_xor(d, off, 32);
        n += __shfl_xor(n, off, 32);
    }
    if (lane == 0) {
        attv[r] = expf(d);
        innv[r] = rsqrtf(fmaxf(n, 1e-12f));
    }
}

// ---------------------------------------------------------------------------
// Main kernel: one wave32 per node.
//  Phase 0: gather the node's 16 object rows (16x128 f32) into LDS, coalesced,
//           with scalar (saddr) base addresses from v_readlane.
//  Phase 1: G = O * R^T via 32 chained V_WMMA_F32_16X16X4_F32; diag(G) = dots.
//  Phase 2: num = sum_e w1[e]*O[e,:] + w2[e]*R[e,:];  out = num / den.
// ---------------------------------------------------------------------------
__global__ __launch_bounds__(WPB * 32) void gat_main_kernel(
    const int*   __restrict__ triples,   // (N_EDGES, 3) int32: subj, rel, obj
    const float* __restrict__ features,  // (N_NODES, 128)
    const float* __restrict__ rel_emb,   // (N_RELS, 128)
    const float* __restrict__ attv,
    const float* __restrict__ innv,
    float*       __restrict__ out,       // (N_NODES, 128)
    int n_nodes)
{
    __shared__ float ldsO[WPB * DEG * ROWP];   // 4 * 16 * 132 * 4B = 33792 B

    const int wid  = (int)(threadIdx.x >> 5);
    const int lane = (int)(threadIdx.x & 31);
    const int node = (int)blockIdx.x * WPB + wid;
    if (node >= n_nodes) return;               // wave-uniform: EXEC stays all-1s

    float* tile = ldsO + wid * (DEG * ROWP);

    // Per-lane edge metadata: lanes l and l+16 both own edge (l & 15).
    const int el = lane & 15;
    const long long ebase = (long long)node * DEG + el;
    const int rel = triples[ebase * 3 + 1];
    const int obj = triples[ebase * 3 + 2];
    const float att = attv[rel];
    const float inn = innv[rel];
    const float* rrow = rel_emb + (size_t)rel * DDIM;

    // ---- Phase 0: stage object rows into LDS (512B coalesced per row) ----
    #pragma unroll
    for (int e = 0; e < DEG; ++e) {
        const int objE = bcast_i(obj, e);      // SGPR -> scalar base address
        const float4 v = *(const float4*)(features + (size_t)objE * DDIM + lane * 4);
        *(float4*)(tile + e * ROWP + lane * 4) = v;
    }
    asm volatile("" ::: "memory");             // keep DS reads after DS writes

    // ---- Phase 1: 16x16 = (16x128) x (128x16) in K=4 f32 WMMA steps ----
    // A frag (16x4 f32): lane l holds O[l&15][kk + koff .. +1], koff = 2*(l>=16)
    // B frag (4x16 f32): lane l holds R[l&15][kk + koff .. +1]  (same pattern)
    const int koff = (lane >> 4) << 1;
    const float* arow = tile + el * ROWP + koff;
    const float* brow = rrow + koff;
    v8f c = {0.f, 0.f, 0.f, 0.f, 0.f, 0.f, 0.f, 0.f};
    #pragma unroll
    for (int kk = 0; kk < DDIM; kk += 4) {
        const v2f a = *(const v2f*)(arow + kk);   // ds_load (2addr-fused), conflict-free
        const v2f b = *(const v2f*)(brow + kk);   // global_load_b64, L2-hot 1MB table
        c = __builtin_amdgcn_wmma_f32_16x16x4_f32(
                false, a, false, b, (short)0, c, false, false);
    }

    // diag(G)[e]: e<8 -> lane e, c[e];  e>=8 -> lane e+16, c[e-8].
    // Both cases: providing lane holds c[lane & 7].
    const int s = lane & 7;
    float md = c[0];
    if (s == 1) md = c[1];
    if (s == 2) md = c[2];
    if (s == 3) md = c[3];
    if (s == 4) md = c[4];
    if (s == 5) md = c[5];
    if (s == 6) md = c[6];
    if (s == 7) md = c[7];

    // ---- Phase 2: weighted accumulate, 4 dims per lane ----
    float ax = 0.f, ay = 0.f, az = 0.f, aw = 0.f;
    float den = 0.f;
    #pragma unroll
    for (int e = 0; e < DEG; ++e) {
        const float w1  = bcast_f(att, e);                   // scalar broadcasts
        const float ine = bcast_f(inn, e);
        const float dt  = bcast_f(md, (e < 8) ? e : (e + 16));
        const float w2  = -2.0f * w1 * ine * dt;             // att * (-2 * dot * inv_norm)
        const float4 o4 = *(const float4*)(tile + e * ROWP + lane * 4);
        const int relE  = bcast_i(rel, e);                   // SGPR -> saddr-form load
        const float4 r4 = *(const float4*)(rel_emb + (size_t)relE * DDIM + lane * 4);
        ax += w1 * o4.x + w2 * r4.x;
        ay += w1 * o4.y + w2 * r4.y;
        az += w1 * o4.z + w2 * r4.z;
        aw += w1 * o4.w + w2 * r4.w;
        den += w1;
    }
    const float id = 1.0f / den;               // den = sum(exp(.)) > 0
    v4f o;
    o.x = ax * id; o.y = ay * id; o.z = az * id; o.w = aw * id;
    // Output written once and never re-read: bypass-ish store to spare L2 for features.
    __builtin_nontemporal_store(o, (v4f*)(out + (size_t)node * DDIM + lane * 4));
}

// ---------------------------------------------------------------------------
extern "C" void kernel_launch(void* const* d_in, const int* in_sizes, int n_in,
                              void* d_out, int out_size, void* d_ws, size_t ws_size,
                              hipStream_t stream) {
    const int*   triples     = (const int*)  d_in[0];  // (1, N_EDGES, 3) int32
    const float* features    = (const float*)d_in[1];  // (N_NODES, 128)
    const float* rel_emb     = (const float*)d_in[2];  // (N_RELS, 128)
    const float* attn_kernel = (const float*)d_in[3];  // (128, 1)
    float*       out         = (float*)d_out;

    const int n_rels  = in_sizes[2] / DDIM;            // 2000
    const int n_nodes = in_sizes[1] / DDIM;            // 100000

    float* attv = (float*)d_ws;
    float* innv = attv + 4096;                          // comfortably past n_rels

    relpre_kernel<<<(n_rels + 7) / 8, 256, 0, stream>>>(
        rel_emb, attn_kernel, attv, innv, n_rels);

    const int blocks = (n_nodes + WPB - 1) / WPB;
    gat_main_kernel<<<blocks, WPB * 32, 0, stream>>>(
        triples, features, rel_emb, attv, innv, out, n_nodes);
}